// Model_59004260712679
// MI455X (gfx1250) — compile-verified
//
#include <hip/hip_runtime.h>
#include <hip/hip_bf16.h>
#include <stdint.h>

// ---------------------------------------------------------------------------
// SAUCD simp_wt scoring: score[b,m] = clip( sum_l |w|[bin(lmbd)] * N * area, 0, 2 )
// Shapes: weight[64], N[8,16,131072], sorted_lmbd[8,16,131073] (use [...,1:]),
//         areas[8,16,131072], out[128].
// HBM-bound streaming reduction (192 MiB read / 23.3 TB/s ~ 8.3 us floor):
// 4-deep async global->LDS pipeline (ASYNCcnt), wave32 reduction,
// deterministic split-K two-pass.
// ---------------------------------------------------------------------------

#define AS1 __attribute__((address_space(1)))
#define AS3 __attribute__((address_space(3)))

typedef int v4i __attribute__((vector_size(16)));

constexpr int kB      = 8;
constexpr int kM      = 16;
constexpr int kL      = 131072;
constexpr int kRows   = kB * kM;          // 128 output cells
constexpr int kSplits = 8;                // split-K factor -> 1024 blocks
constexpr int kChunk  = kL / kSplits;     // 16384 elems per block
constexpr int kTPB    = 256;              // 8 waves (wave32)
constexpr int kVec    = 4;                // float4 per lane per stream per iter
constexpr int kIterEl = kTPB * kVec;      // 1024 elems per iter
constexpr int kIters  = kChunk / kIterEl; // 16
constexpr int kWaves  = kTPB / 32;
constexpr int kModel  = 64;
constexpr int kBufs   = 4;                // async pipeline depth (3 ahead)

// ---- CDNA5 async global->LDS copy (ASYNCcnt-tracked), with asm fallback ----
__device__ __forceinline__ void async_cp_b128(const float* g, float* l) {
#if defined(__HIP_DEVICE_COMPILE__)
#if __has_builtin(__builtin_amdgcn_global_load_async_to_lds_b128)
  __builtin_amdgcn_global_load_async_to_lds_b128(
      (AS1 v4i*)const_cast<float*>(g), (AS3 v4i*)l, /*offset=*/0, /*cpol=*/0);
#else
  uint32_t loff = (uint32_t)(uintptr_t)(AS3 char*)l;   // 32-bit LDS byte address
  uint64_t ga   = (uint64_t)(uintptr_t)g;              // 64-bit global address
  asm volatile("global_load_async_to_lds_b128 %0, %1, off"
               :: "v"(loff), "v"(ga) : "memory");
#endif
#endif
}

#if defined(__HIP_DEVICE_COMPILE__)
#if __has_builtin(__builtin_amdgcn_s_wait_asynccnt)
#define WAIT_ASYNC(n) __builtin_amdgcn_s_wait_asynccnt(n)
#else
#define WAIT_ASYNC(n) asm volatile("s_wait_asynccnt %0" :: "i"(n) : "memory")
#endif
#else
#define WAIT_ASYNC(n) do {} while (0)
#endif

// log-lambda bin -> |weight| lookup (LUT lives in LDS)
__device__ __forceinline__ float seg_weight(const float* lut, float l) {
  float ll  = __logf(fmaxf(l, 1e-8f));                 // v_log_f32 (trans)
  int   idx = (int)floorf((ll + 6.0f) * (1.0f / 0.105f));
  idx = idx < 0 ? 0 : (idx > kModel - 1 ? kModel - 1 : idx);
  return lut[idx];                                     // ds_load_b32
}

__global__ __launch_bounds__(kTPB) void saucd_partial(
    const float* __restrict__ weight,
    const float* __restrict__ Narr,
    const float* __restrict__ lmbd,
    const float* __restrict__ areas,
    float* __restrict__ partial) {
  __shared__ __align__(16) float stage[kBufs][3][kIterEl]; // 48 KB, 4-deep
  __shared__ float lut[kModel];
  __shared__ float wsum[kWaves];

  const int tid   = threadIdx.x;
  const int split = blockIdx.x;        // 0..kSplits-1
  const int row   = blockIdx.y;        // 0..127  (b*16+m)

  if (tid < kModel) lut[tid] = fabsf(weight[tid]);
  __syncthreads();

  const size_t base = (size_t)row * kL + (size_t)split * kChunk;           // N / areas
  const size_t lb   = (size_t)row * (kL + 1) + 1 + (size_t)split * kChunk; // lmbd[...,1:]
  const int slot = tid * kVec;

  float acc = 0.0f;

  // Per-wave LDS slots: ASYNCcnt alone orders producer/consumer (completion is
  // in-order per wave), so no block barrier inside the pipeline loop.
#pragma unroll
  for (int p = 0; p < kBufs - 1; ++p) {  // prologue: buffers 0..2 in flight
    const int off = p * kIterEl + slot;
    async_cp_b128(Narr  + base + off, &stage[p][0][slot]);
    async_cp_b128(areas + base + off, &stage[p][1][slot]);
    async_cp_b128(lmbd  + lb   + off, &stage[p][2][slot]);
  }

  for (int i = 0; i < kIters; ++i) {
    const int buf = i & (kBufs - 1);
    if (i + kBufs - 1 < kIters) {
      const int nb  = (i + kBufs - 1) & (kBufs - 1);
      const int off = (i + kBufs - 1) * kIterEl + slot;
      async_cp_b128(Narr  + base + off, &stage[nb][0][slot]);
      async_cp_b128(areas + base + off, &stage[nb][1][slot]);
      async_cp_b128(lmbd  + lb   + off, &stage[nb][2][slot]);
      WAIT_ASYNC(6);                 // newest 2 buffers may be in flight
    } else if (i + 2 < kIters) {
      WAIT_ASYNC(6);                 // drain: 2 buffers still ahead
    } else if (i + 1 < kIters) {
      WAIT_ASYNC(3);                 // drain: 1 buffer still ahead
    } else {
      WAIT_ASYNC(0);
    }
    asm volatile("" ::: "memory");   // keep LDS reads below the wait

    const float4 n4 = *reinterpret_cast<const float4*>(&stage[buf][0][slot]);
    const float4 a4 = *reinterpret_cast<const float4*>(&stage[buf][1][slot]);
    const float4 l4 = *reinterpret_cast<const float4*>(&stage[buf][2][slot]);

    acc += seg_weight(lut, l4.x) * n4.x * a4.x;
    acc += seg_weight(lut, l4.y) * n4.y * a4.y;
    acc += seg_weight(lut, l4.z) * n4.z * a4.z;
    acc += seg_weight(lut, l4.w) * n4.w * a4.w;
  }

  // wave32 reduction, then cross-wave via LDS
  for (int off = 16; off > 0; off >>= 1) acc += __shfl_down(acc, off, 32);
  if ((tid & 31) == 0) wsum[tid >> 5] = acc;
  __syncthreads();
  if (tid == 0) {
    float s = 0.0f;
#pragma unroll
    for (int w = 0; w < kWaves; ++w) s += wsum[w];
    partial[row * kSplits + split] = s;
  }
}

__global__ void saucd_finalize(const float* __restrict__ partial,
                               float* __restrict__ out) {
  const int r = threadIdx.x;
  if (r < kRows) {
    float s = 0.0f;
#pragma unroll
    for (int k = 0; k < kSplits; ++k) s += partial[r * kSplits + k];
    out[r] = fminf(fmaxf(s, 0.0f), 2.0f);
  }
}

extern "C" void kernel_launch(void* const* d_in, const int* in_sizes, int n_in,
                              void* d_out, int out_size, void* d_ws, size_t ws_size,
                              hipStream_t stream) {
  (void)in_sizes; (void)n_in; (void)out_size; (void)ws_size;
  const float* weight = (const float*)d_in[0];
  const float* Narr   = (const float*)d_in[1];
  const float* lmbd   = (const float*)d_in[2];
  const float* areas  = (const float*)d_in[3];
  float* out     = (float*)d_out;
  float* partial = (float*)d_ws;      // kRows*kSplits floats = 4 KB scratch

  dim3 grid(kSplits, kRows);          // 1024 blocks x 256 threads (8 waves)
  saucd_partial<<<grid, dim3(kTPB), 0, stream>>>(weight, Narr, lmbd, areas, partial);
  saucd_finalize<<<dim3(1), dim3(kRows), 0, stream>>>(partial, out);
}